// NeRF_29781303230883
// MI455X (gfx1250) — compile-verified
//
#include <hip/hip_runtime.h>
#include <cstdint>

// ---------------------------------------------------------------------------
// NeRF volume rendering for MI455X (gfx1250, wave32).
// One wave per ray, one lane per sample (NUM_SAMPLES == 32 == wave32).
// Predictions staged via async global->LDS copies (ASYNCcnt pipeline);
// scan + reductions done with wave32 shuffles / packed-f32 adds.
// Memory-roofline kernel: ~420 MB traffic -> ~18 us at 23.3 TB/s.
// ---------------------------------------------------------------------------

#define NS 32              // samples per ray == wave32 lanes
#define WAVES_PER_BLOCK 8  // 256 threads
#define RPW 4              // rays processed per wave (pipeline length)

#if __has_builtin(__builtin_amdgcn_global_load_async_to_lds_b128)
#define HAVE_ASYNC_BUILTIN 1
#endif

typedef int v4i __attribute__((ext_vector_type(4)));
typedef __attribute__((address_space(1))) v4i gv4i_t;  // global int4
typedef __attribute__((address_space(3))) v4i lv4i_t;  // LDS int4
typedef __attribute__((address_space(3))) void lv_t;

// Async copy of one 16B element per lane: 32 lanes -> 512B contiguous tile.
__device__ __forceinline__ void async_copy_b128(const void* g, void* l) {
#ifdef HAVE_ASYNC_BUILTIN
  __builtin_amdgcn_global_load_async_to_lds_b128((gv4i_t*)g, (lv4i_t*)l, 0, 0);
#else
  unsigned lds_addr = (unsigned)(uintptr_t)(lv_t*)l;
  unsigned long long gaddr = (unsigned long long)(uintptr_t)g;
  asm volatile("global_load_async_to_lds_b128 %0, %1, off"
               :: "v"(lds_addr), "v"(gaddr)
               : "memory");
#endif
}

__device__ __forceinline__ void wait_async_le1() {
  asm volatile("s_wait_asynccnt 1" ::: "memory");
}
__device__ __forceinline__ void wait_async_le0() {
  asm volatile("s_wait_asynccnt 0" ::: "memory");
}
__device__ __forceinline__ void wait_ds0() {
  asm volatile("s_wait_dscnt 0" ::: "memory");
}

__device__ __forceinline__ float fast_sigmoid(float x) {
  return 1.0f / (1.0f + __expf(-x));
}

// Per-ray volume rendering: lane == sample index.
__device__ __forceinline__ void render_ray(float4 p, float t, int lane, int ray,
                                           float* __restrict__ out_rgb,
                                           float* __restrict__ out_depth) {
  // delta_n = t_{n+1} - t_n, last = 1e10
  float tn    = __shfl_down(t, 1, 32);
  float delta = (lane == NS - 1) ? 1e10f : (tn - t);

  float sigma = fmaxf(p.w, 0.0f);                 // relu
  float alpha = 1.0f - __expf(-sigma * delta);
  float et    = (1.0f - alpha) + 1e-10f;          // exp_term + eps

  // Inclusive cumprod across lanes (Hillis-Steele, 5 steps on wave32).
  float cp = et;
#pragma unroll
  for (int off = 1; off < NS; off <<= 1) {
    float o = __shfl_up(cp, off, 32);
    if (lane >= off) cp *= o;
  }
  // Exclusive: transmittance[0]=1, transmittance[n]=cp[n-1]
  float T = __shfl_up(cp, 1, 32);
  if (lane == 0) T = 1.0f;

  float w = alpha * T;
  // Two packed accumulators: (rgb0, rgb1) and (rgb2, depth) -> V_PK_ADD_F32.
  float2 a = make_float2(w * fast_sigmoid(p.x), w * fast_sigmoid(p.y));
  float2 b = make_float2(w * fast_sigmoid(p.z), w * t);

  // Butterfly sum across the 32 samples.
#pragma unroll
  for (int m = 16; m >= 1; m >>= 1) {
    float2 as = make_float2(__shfl_xor(a.x, m, 32), __shfl_xor(a.y, m, 32));
    float2 bs = make_float2(__shfl_xor(b.x, m, 32), __shfl_xor(b.y, m, 32));
    a += as;
    b += bs;
  }

  if (lane == 0) {
    size_t r3 = (size_t)ray * 3;
    out_rgb[r3 + 0] = a.x;
    out_rgb[r3 + 1] = a.y;
    out_rgb[r3 + 2] = b.x;
    out_depth[ray]  = b.y;
  }
}

__global__ __launch_bounds__(WAVES_PER_BLOCK * 32, 1)
void nerf_render_kernel(const float4* __restrict__ pred,   // [R, NS] float4
                        const float*  __restrict__ tvals,  // [R, NS]
                        float* __restrict__ out_rgb,       // [R, 3]
                        float* __restrict__ out_depth,     // [R]
                        int R) {
  __shared__ float4 sbuf[WAVES_PER_BLOCK][2][NS];  // 8 KB: per-wave double buffer

  const int lane = threadIdx.x & 31;
  const int wave = threadIdx.x >> 5;
  const int ray0 = (blockIdx.x * WAVES_PER_BLOCK + wave) * RPW;
  if (ray0 >= R) return;

  if (ray0 + RPW <= R) {
    // ---------------- Fast path: full tile, branch-free pipeline ----------
    async_copy_b128(&pred[(size_t)ray0 * NS + lane], &sbuf[wave][0][lane]);
    float t_cur = tvals[(size_t)ray0 * NS + lane];

#pragma unroll
    for (int i = 0; i < RPW; ++i) {
      const int rcur = ray0 + i;
      float t_next = 0.0f;
      if (i + 1 < RPW) {
        // Buffer (i+1)&1 was ds_read last iteration; its reads have retired
        // (data consumed by VALU), but fence explicitly: async LDS writes are
        // unordered vs DS ops (ISA Flow&Sync 7.3).
        wait_ds0();
        async_copy_b128(&pred[(size_t)(rcur + 1) * NS + lane],
                        &sbuf[wave][(i + 1) & 1][lane]);
        t_next = tvals[(size_t)(rcur + 1) * NS + lane];
      }
      // Async loads complete in order: <=1 outstanding means the current
      // buffer's copy has landed in LDS.
      if (i + 1 < RPW) wait_async_le1(); else wait_async_le0();

      float4 p = sbuf[wave][i & 1][lane];  // ds_load_b128, conflict-free
      render_ray(p, t_cur, lane, rcur, out_rgb, out_depth);
      t_cur = t_next;
    }
  } else {
    // ---------------- Tail path: per-ray guarded, direct loads ------------
    for (int i = 0; i < RPW; ++i) {
      const int rcur = ray0 + i;
      if (rcur >= R) break;
      float4 p   = pred[(size_t)rcur * NS + lane];   // coalesced b128
      float  t_c = tvals[(size_t)rcur * NS + lane];  // coalesced b32
      render_ray(p, t_c, lane, rcur, out_rgb, out_depth);
    }
  }
}

extern "C" void kernel_launch(void* const* d_in, const int* in_sizes, int n_in,
                              void* d_out, int out_size, void* d_ws, size_t ws_size,
                              hipStream_t stream) {
  (void)n_in; (void)d_ws; (void)ws_size; (void)out_size;

  const float4* pred  = (const float4*)d_in[0];  // [B,H,W,NS,4] fp32
  const float*  tvals = (const float*)d_in[1];   // [B,H,W,NS]   fp32
  const int R = in_sizes[1] / NS;                // number of rays (B*H*W)

  float* out       = (float*)d_out;
  float* out_rgb   = out;                        // [R,3]
  float* out_depth = out + (size_t)R * 3;        // [R]

  const int wavesNeeded = (R + RPW - 1) / RPW;
  const int blocks      = (wavesNeeded + WAVES_PER_BLOCK - 1) / WAVES_PER_BLOCK;

  hipLaunchKernelGGL(nerf_render_kernel, dim3(blocks), dim3(WAVES_PER_BLOCK * 32),
                     0, stream, pred, tvals, out_rgb, out_depth, R);
}